// SubEquiMLP_27986006901349
// MI455X (gfx1250) — compile-verified
//
#include <hip/hip_runtime.h>

typedef __attribute__((ext_vector_type(2))) float v2f;
typedef __attribute__((ext_vector_type(8))) float v8f;

namespace {
constexpr int kB    = 4;
constexpr int kN    = 512;
constexpr int kD    = 64;
constexpr int kDH   = 63;
constexpr int kHID  = 64;
constexpr int kRows = kB * kN;           // 2048 MLP rows total
constexpr int kTileM = 16;               // WMMA M tile
constexpr int kWaves = 2;                // waves per block
constexpr int kThreads = kWaves * 32;    // wave32
constexpr int kActStride = 68;           // 64 + pad (LDS bank-conflict avoidance)
}

// zsum[b][d] = sum_m z[b, m, d]   (B*D = 256 threads, one block)
__global__ void zsum_kernel(const float* __restrict__ z, float* __restrict__ zsum) {
  int t = threadIdx.x;                   // 0..255
  int b = t >> 6;
  int d = t & 63;
  const float* p = z + (size_t)b * kN * kD + d;
  float s = 0.f;
  for (int m = 0; m < kN; ++m) s += p[(size_t)m * kD];
  zsum[t] = s;
}

// One wave computes one 16-row tile of the diagonal MLP, all four 64x64
// layers via V_WMMA_F32_16X16X4_F32, then emits the rank-1 update
// out[b,n,:] = (sum_o y[n,o]) * zsum[b,:].
__global__ __launch_bounds__(kThreads)
void mlp_diag_kernel(const float* __restrict__ z, const float* __restrict__ h,
                     const float* __restrict__ g,
                     const float* __restrict__ W0, const float* __restrict__ B0,
                     const float* __restrict__ W1, const float* __restrict__ B1,
                     const float* __restrict__ W2, const float* __restrict__ B2,
                     const float* __restrict__ W3, const float* __restrict__ B3,
                     const float* __restrict__ zsum, float* __restrict__ out)
{
  __shared__ float sW[kHID * kHID];                     // current layer weights (16 KB)
  __shared__ float sBias[kHID];
  __shared__ float sAct[kWaves * kTileM * kActStride];  // per-wave activation tiles

  const int tid  = threadIdx.x;
  const int wave = tid >> 5;
  const int lane = tid & 31;
  const int half = lane >> 4;            // 0: lanes 0-15, 1: lanes 16-31
  const int n16  = lane & 15;

  const int tile = blockIdx.x * kWaves + wave;
  const int row0 = tile * kTileM;        // 512 % 16 == 0 -> tile stays in one batch
  const int bidx = row0 / kN;

  float* act = sAct + wave * (kTileM * kActStride);     // [16][kActStride]

  // ---- build MLP input X[m,:] = [ ||z_row||^2 + g.g , h_row(63) ] ----
  const float gg = g[0] * g[0] + g[1] * g[1] + g[2] * g[2];
  {
    const float* zr = z + (size_t)(row0 + n16) * kD + half * 32;
    float ss = 0.f;
#pragma unroll
    for (int k = 0; k < 32; ++k) { float v = zr[k]; ss += v * v; }
    ss += __shfl_xor(ss, 16, 32);        // combine the two half-row partials
    if (half == 0) act[n16 * kActStride] = ss + gg;
  }
  for (int t = lane; t < kTileM * kDH; t += 32) {
    int m = t / kDH;
    int j = t - m * kDH;
    act[m * kActStride + 1 + j] = h[(size_t)(row0 + m) * kDH + j];
  }

  const float* Ws[4] = {W0, W1, W2, W3};
  const float* Bs[4] = {B0, B1, B2, B3};

#pragma unroll
  for (int layer = 0; layer < 4; ++layer) {
    __syncthreads();                     // prior layer's sW reads done; X build done
    for (int t = tid; t < kHID * kHID; t += kThreads) sW[t] = Ws[layer][t];
    if (tid < kHID) sBias[tid] = Bs[layer][tid];
    __syncthreads();

    // C init = bias broadcast (C/D layout: row = r + 8*half, col = n16)
    v8f acc[4];
#pragma unroll
    for (int nt = 0; nt < 4; ++nt) {
      float bv = sBias[nt * 16 + n16];
#pragma unroll
      for (int r = 0; r < 8; ++r) acc[nt][r] = bv;
    }

    // K loop: 16 steps of K=4 f32 WMMA per output tile
#pragma unroll
    for (int kk = 0; kk < kHID; kk += 4) {
      // A 16x4 layout: lane = row M, VGPR0/1 hold K = 2*half, 2*half+1
      v2f a;
      a.x = act[n16 * kActStride + kk + 2 * half + 0];
      a.y = act[n16 * kActStride + kk + 2 * half + 1];
#pragma unroll
      for (int nt = 0; nt < 4; ++nt) {
        // B 4x16: B[k', n] = W[o, kk + k'] with o = nt*16 + n (x @ W^T)
        v2f bf;
        const int o = nt * 16 + n16;
        bf.x = sW[o * kHID + kk + 2 * half + 0];
        bf.y = sW[o * kHID + kk + 2 * half + 1];
        acc[nt] = __builtin_amdgcn_wmma_f32_16x16x4_f32(
            false, a, false, bf, (short)0, acc[nt], false, false);
      }
    }

    // ReLU (layers 0-2) + writeback in C/D layout -> LDS (becomes next A)
    const bool last = (layer == 3);
#pragma unroll
    for (int nt = 0; nt < 4; ++nt) {
#pragma unroll
      for (int r = 0; r < 8; ++r) {
        float v = acc[nt][r];
        if (!last) v = fmaxf(v, 0.f);
        act[(r + 8 * half) * kActStride + nt * 16 + n16] = v;
      }
    }
  }

  // out[b, row, d] = (sum_o y[row, o]) * zsum[b, d]
  // lane pair (L, L+16) handles row n16; each covers 32 of the 64 d's.
  float s = 0.f;
#pragma unroll
  for (int k = 0; k < kHID; ++k) s += act[n16 * kActStride + k];
  const float* zs = zsum + bidx * kD + half * 32;
  float* op = out + (size_t)(row0 + n16) * kD + half * 32;
#pragma unroll
  for (int d = 0; d < 32; ++d) op[d] = s * zs[d];
}

extern "C" void kernel_launch(void* const* d_in, const int* in_sizes, int n_in,
                              void* d_out, int out_size, void* d_ws, size_t ws_size,
                              hipStream_t stream) {
  const float* z  = (const float*)d_in[0];
  const float* h  = (const float*)d_in[1];
  const float* g  = (const float*)d_in[2];
  const float* W0 = (const float*)d_in[3];
  const float* b0 = (const float*)d_in[4];
  const float* W1 = (const float*)d_in[5];
  const float* b1 = (const float*)d_in[6];
  const float* W2 = (const float*)d_in[7];
  const float* b2 = (const float*)d_in[8];
  const float* W3 = (const float*)d_in[9];
  const float* b3 = (const float*)d_in[10];
  float* out  = (float*)d_out;
  float* zsum = (float*)d_ws;            // 256 floats of scratch

  zsum_kernel<<<1, kB * kD, 0, stream>>>(z, zsum);
  mlp_diag_kernel<<<kRows / (kTileM * kWaves), kThreads, 0, stream>>>(
      z, h, g, W0, b0, W1, b1, W2, b2, W3, b3, zsum, out);
}